// HyperLayer_83425444758105
// MI455X (gfx1250) — compile-verified
//
#include <hip/hip_runtime.h>

#define D_IN   4096
#define D_OUT  4096
#define BCOLS  64
#define NPTS   262144
#define KSPLIT 4

typedef __attribute__((ext_vector_type(2))) float v2f;
typedef __attribute__((ext_vector_type(8))) float v8f;

__device__ __forceinline__ void atomAddF(float* p, float v) {
    // lowers to global_atomic_add_f32 (no CAS loop) on gfx1250
    unsafeAtomicAdd(p, v);
}

// ---------------------------------------------------------------------------
// Pass 0: zero the dense S^T scratch (d_ws) and the output accumulator.
// ---------------------------------------------------------------------------
__global__ void zero_kernel(float4* __restrict__ s4, float4* __restrict__ o4,
                            int ns4, int no4) {
    int idx    = blockIdx.x * blockDim.x + threadIdx.x;
    int stride = gridDim.x * blockDim.x;
    float4 z; z.x = 0.f; z.y = 0.f; z.z = 0.f; z.w = 0.f;
    for (int i = idx; i < ns4; i += stride) s4[i] = z;
    for (int i = idx; i < no4; i += stride) o4[i] = z;
}

// ---------------------------------------------------------------------------
// Pass 1: bilinear discretization -> scatter weights into dense ST[r][c]
// (S transposed: row = input index r, col = output index c).
// Exactly mirrors reference: all 4 floor/ceil corners are added, with
// prop = prod(1 - |corner - ind|).
// ---------------------------------------------------------------------------
__global__ void scatter_kernel(const float2* __restrict__ ind,
                               const float*  __restrict__ val,
                               float* __restrict__ ST, int n) {
    int i = blockIdx.x * blockDim.x + threadIdx.x;
    if (i >= n) return;
    float2 p = ind[i];
    float v  = val[i];
    float f0 = floorf(p.x), c0 = ceilf(p.x);
    float f1 = floorf(p.y), c1 = ceilf(p.y);
    float wf0 = 1.0f - (p.x - f0), wc0 = 1.0f - (c0 - p.x);
    float wf1 = 1.0f - (p.y - f1), wc1 = 1.0f - (c1 - p.y);
    int r0 = (int)f0, r1 = (int)c0;
    int cA = (int)f1, cB = (int)c1;
    atomAddF(ST + r0 * D_OUT + cA, v * wf0 * wf1);
    atomAddF(ST + r0 * D_OUT + cB, v * wf0 * wc1);
    atomAddF(ST + r1 * D_OUT + cA, v * wc0 * wf1);
    atomAddF(ST + r1 * D_OUT + cB, v * wc0 * wc1);
}

// ---------------------------------------------------------------------------
// Pass 2: out[m, n] = sum_k ST[k][m] * x[k][n]  via V_WMMA_F32_16X16X4_F32.
// One wave32 per (16-row strip, K-chunk); 4 N-tiles (all 64 cols) per wave.
// grid = (D_OUT/16)*KSPLIT waves = 1024 waves = 128 blocks x 8 waves.
// ---------------------------------------------------------------------------
__global__ void wmma_gemm_kernel(const float* __restrict__ ST,
                                 const float* __restrict__ x,
                                 float* __restrict__ out) {
    const int lane   = threadIdx.x & 31;
    const int wave   = blockIdx.x * (blockDim.x >> 5) + (threadIdx.x >> 5);
    const int strip  = wave / KSPLIT;   // 0..255  -> out rows [strip*16, +16)
    const int kchunk = wave % KSPLIT;   // 0..3    -> K range of 1024
    const int m   = strip * 16 + (lane & 15);
    const int kk  = (lane >> 4) << 1;   // lanes 16-31 carry K+2, K+3
    const int n15 = lane & 15;

    v8f acc0 = {}, acc1 = {}, acc2 = {}, acc3 = {};

    const int kbeg = kchunk * (D_IN / KSPLIT);
    const int kend = kbeg + (D_IN / KSPLIT);
    #pragma unroll 4
    for (int k = kbeg; k < kend; k += 4) {
        const int kr = k + kk;
        v2f a;
        a.x = ST[kr * D_OUT + m];
        a.y = ST[(kr + 1) * D_OUT + m];

        const float* xr0 = x + kr * BCOLS;
        const float* xr1 = x + (kr + 1) * BCOLS;
        v2f b0, b1, b2, b3;
        b0.x = xr0[n15];      b0.y = xr1[n15];
        b1.x = xr0[16 + n15]; b1.y = xr1[16 + n15];
        b2.x = xr0[32 + n15]; b2.y = xr1[32 + n15];
        b3.x = xr0[48 + n15]; b3.y = xr1[48 + n15];

        acc0 = __builtin_amdgcn_wmma_f32_16x16x4_f32(false, a, false, b0, (short)0, acc0, false, false);
        acc1 = __builtin_amdgcn_wmma_f32_16x16x4_f32(false, a, false, b1, (short)0, acc1, false, false);
        acc2 = __builtin_amdgcn_wmma_f32_16x16x4_f32(false, a, false, b2, (short)0, acc2, false, false);
        acc3 = __builtin_amdgcn_wmma_f32_16x16x4_f32(false, a, false, b3, (short)0, acc3, false, false);
    }

    // C/D 16x16 layout: VGPR i -> row (i + 8*(lane>>4)), col (lane&15)
    const int rbase = strip * 16 + ((lane >> 4) << 3);
    #pragma unroll
    for (int i = 0; i < 8; ++i) {
        float* orow = out + (rbase + i) * BCOLS + n15;
        atomAddF(orow +  0, acc0[i]);
        atomAddF(orow + 16, acc1[i]);
        atomAddF(orow + 32, acc2[i]);
        atomAddF(orow + 48, acc3[i]);
    }
}

// ---------------------------------------------------------------------------
// Fallback (only if d_ws can't hold the 64MB dense S): direct atomic scatter.
// ---------------------------------------------------------------------------
__global__ void fallback_scatter(const float2* __restrict__ ind,
                                 const float*  __restrict__ val,
                                 const float*  __restrict__ x,
                                 float* __restrict__ out, int n) {
    int i = blockIdx.x * blockDim.x + threadIdx.x;
    if (i >= n) return;
    float2 p = ind[i];
    float v  = val[i];
    float f0 = floorf(p.x), c0 = ceilf(p.x);
    float f1 = floorf(p.y), c1 = ceilf(p.y);
    float wf0 = 1.0f - (p.x - f0), wc0 = 1.0f - (c0 - p.x);
    float wf1 = 1.0f - (p.y - f1), wc1 = 1.0f - (c1 - p.y);
    int r0 = (int)f0, r1 = (int)c0;
    int cA = (int)f1, cB = (int)c1;
    for (int b = 0; b < BCOLS; ++b) {
        float xr = wf0 * x[r0 * BCOLS + b] + wc0 * x[r1 * BCOLS + b];
        atomAddF(&out[cA * BCOLS + b], v * wf1 * xr);
        atomAddF(&out[cB * BCOLS + b], v * wc1 * xr);
    }
}

extern "C" void kernel_launch(void* const* d_in, const int* in_sizes, int n_in,
                              void* d_out, int out_size, void* d_ws, size_t ws_size,
                              hipStream_t stream) {
    const float2* ind = (const float2*)d_in[0];   // [N,2] f32
    const float*  val = (const float*)d_in[1];    // [N]   f32
    const float*  x   = (const float*)d_in[2];    // [D_IN, B] f32
    float* out = (float*)d_out;                   // [D_OUT, B] f32
    const int n = in_sizes[1];                    // N

    const size_t sBytes = (size_t)D_IN * D_OUT * sizeof(float);  // 64 MB
    const int no4 = (D_OUT * BCOLS) / 4;

    if (ws_size >= sBytes) {
        float* ST = (float*)d_ws;
        const int ns4 = (D_IN * D_OUT) / 4;
        zero_kernel<<<2048, 256, 0, stream>>>((float4*)ST, (float4*)out, ns4, no4);
        scatter_kernel<<<(n + 255) / 256, 256, 0, stream>>>(ind, val, ST, n);
        const int gemmBlocks = (D_OUT / 16) * KSPLIT / 8;  // 128
        wmma_gemm_kernel<<<gemmBlocks, 256, 0, stream>>>(ST, x, out);
    } else {
        zero_kernel<<<512, 256, 0, stream>>>(nullptr, (float4*)out, 0, no4);
        fallback_scatter<<<(n + 255) / 256, 256, 0, stream>>>(ind, val, x, out, n);
    }
}